// GraphEmbedding_60773787239061
// MI455X (gfx1250) — compile-verified
//
#include <hip/hip_runtime.h>
#include <hip/hip_bf16.h>
#include <math.h>

// ---------------------------------------------------------------------------
// GATv2 layer for MI455X (gfx1250, wave32).
// Dense GEMMs via V_WMMA_F32_16X16X4_F32 (full fp32 precision).
// Edge phase: on-the-fly edge-embedding + segment softmax with float atomics.
// ---------------------------------------------------------------------------

typedef float v2f __attribute__((ext_vector_type(2)));
typedef float v8f __attribute__((ext_vector_type(8)));

#define DMODEL 128
#define NHEAD  8
#define CHEAD  16
#define EDIM   16
#define SQRT_D 11.313708498984760f   // sqrt(128)

// ---------------- utility: grid-stride fill ----------------
__global__ void k_fill(float* __restrict__ p, float v, size_t count) {
    size_t i = (size_t)blockIdx.x * blockDim.x + threadIdx.x;
    size_t stride = (size_t)gridDim.x * blockDim.x;
    for (; i < count; i += stride) p[i] = v;
}

// ---------------- float atomic max via int/uint ordering trick -------------
__device__ __forceinline__ void atomicMaxF(float* addr, float val) {
    if (!(val < 0.0f)) {
        atomicMax((int*)addr, __float_as_int(val));
    } else {
        atomicMin((unsigned int*)addr, __float_as_uint(val));
    }
}

// ---------------------------------------------------------------------------
// Kernel 1: h = emb[x]*sqrt(D);  xl = h@W_l + b_l;  xr = h@W_r + b_r
// Block: 256 threads (8 waves), 32 rows per block.
// Wave w owns output columns [16w, 16w+16); computes 2 M-tiles for L and R.
// WMMA f32 16x16x4, K-loop over 128 in steps of 4.
// ---------------------------------------------------------------------------
__global__ void k_embed_gemm(const int* __restrict__ x,
                             const float* __restrict__ emb,
                             const float* __restrict__ Wl,
                             const float* __restrict__ bl,
                             const float* __restrict__ Wr,
                             const float* __restrict__ br,
                             float* __restrict__ xl,
                             float* __restrict__ xr,
                             int n)
{
    __shared__ float hs[32][DMODEL + 4];   // padded to dodge bank conflicts

    const int row0 = blockIdx.x * 32;

    // Stage 32 x 128 tile of h = emb[x]*sqrt(D) (zero-pad past n).
    for (int i = threadIdx.x; i < 32 * DMODEL; i += 256) {
        int r = i >> 7, c = i & (DMODEL - 1);
        int row = row0 + r;
        float v = 0.0f;
        if (row < n) v = emb[(size_t)x[row] * DMODEL + c] * SQRT_D;
        hs[r][c] = v;
    }
    __syncthreads();

    const int wave = threadIdx.x >> 5;       // 0..7
    const int lane = threadIdx.x & 31;
    const int half = lane >> 4;              // 0 or 1
    const int r    = lane & 15;
    const int n0   = wave * 16;              // output column tile

    v8f accl0 = {}; v8f accl1 = {};
    v8f accr0 = {}; v8f accr1 = {};

    #pragma unroll 4
    for (int k0 = 0; k0 < DMODEL; k0 += 4) {
        const int ka = k0 + 2 * half;        // lane's K base (v + 2*half)

        // A fragments: A[m = r][k = ka + v], two M-tiles
        v2f a0; a0.x = hs[r][ka];        a0.y = hs[r][ka + 1];
        v2f a1; a1.x = hs[16 + r][ka];   a1.y = hs[16 + r][ka + 1];

        // B fragments: B[k = ka + v][n = n0 + r]
        v2f bL; bL.x = Wl[ka * DMODEL + n0 + r];
                bL.y = Wl[(ka + 1) * DMODEL + n0 + r];
        v2f bR; bR.x = Wr[ka * DMODEL + n0 + r];
                bR.y = Wr[(ka + 1) * DMODEL + n0 + r];

        accl0 = __builtin_amdgcn_wmma_f32_16x16x4_f32(false, a0, false, bL,
                                                      (short)0, accl0, false, false);
        accl1 = __builtin_amdgcn_wmma_f32_16x16x4_f32(false, a1, false, bL,
                                                      (short)0, accl1, false, false);
        accr0 = __builtin_amdgcn_wmma_f32_16x16x4_f32(false, a0, false, bR,
                                                      (short)0, accr0, false, false);
        accr1 = __builtin_amdgcn_wmma_f32_16x16x4_f32(false, a1, false, bR,
                                                      (short)0, accr1, false, false);
    }

    const float bln = bl[n0 + r];
    const float brn = br[n0 + r];

    #pragma unroll
    for (int v = 0; v < 8; v++) {
        int m = v + 8 * half;                // D[m][n0+r] lives in acc[v]
        int rowA = row0 + m;
        if (rowA < n) {
            xl[(size_t)rowA * DMODEL + n0 + r] = accl0[v] + bln;
            xr[(size_t)rowA * DMODEL + n0 + r] = accr0[v] + brn;
        }
        int rowB = row0 + 16 + m;
        if (rowB < n) {
            xl[(size_t)rowB * DMODEL + n0 + r] = accl1[v] + bln;
            xr[(size_t)rowB * DMODEL + n0 + r] = accr1[v] + brn;
        }
    }
}

// ---------------------------------------------------------------------------
// Kernel 2: per-edge scores. One wave per edge; lane owns 4 channels.
// eemb computed on the fly (ew[e,16] @ W_e) to avoid 410MB intermediate.
// score[e,h] = att[h] . leaky_relu(xl[src]+xr[dst]+eemb, 0.2)
// Also atomic-max into smax[dst,h].
// ---------------------------------------------------------------------------
__global__ void k_edge_score(const int* __restrict__ ei,
                             const float* __restrict__ ew,
                             const float* __restrict__ We,
                             const float* __restrict__ att,
                             const float* __restrict__ xl,
                             const float* __restrict__ xr,
                             float* __restrict__ score,
                             float* __restrict__ smax,
                             int E)
{
    __shared__ float sWe[EDIM * DMODEL];   // 8 KB, L2->LDS once per block
    __shared__ float sAtt[DMODEL];
    for (int i = threadIdx.x; i < EDIM * DMODEL; i += blockDim.x) sWe[i] = We[i];
    for (int i = threadIdx.x; i < DMODEL; i += blockDim.x) sAtt[i] = att[i];
    __syncthreads();

    const int e = blockIdx.x * (blockDim.x >> 5) + (threadIdx.x >> 5);
    if (e >= E) return;
    const int lane = threadIdx.x & 31;

    const int src = ei[e];
    const int dst = ei[E + e];

    float w[EDIM];
    #pragma unroll
    for (int k = 0; k < EDIM; k++) w[k] = ew[(size_t)e * EDIM + k];  // wave-broadcast

    const int c0 = lane * 4;
    float part = 0.0f;
    #pragma unroll
    for (int j = 0; j < 4; j++) {
        const int c = c0 + j;
        float s = xl[(size_t)src * DMODEL + c] + xr[(size_t)dst * DMODEL + c];
        #pragma unroll
        for (int k = 0; k < EDIM; k++) s += w[k] * sWe[k * DMODEL + c];
        s = (s > 0.0f) ? s : 0.2f * s;       // leaky_relu, slope 0.2
        part += s * sAtt[c];                 // att flat-[128] matches channel layout
    }
    // reduce the head's 16 channels across its 4 lanes (lanes 4h..4h+3)
    part += __shfl_xor(part, 1, 32);
    part += __shfl_xor(part, 2, 32);
    if ((lane & 3) == 0) {
        const int h = lane >> 2;
        score[(size_t)e * NHEAD + h] = part;
        atomicMaxF(&smax[(size_t)dst * NHEAD + h], part);
    }
}

// ---------------------------------------------------------------------------
// Kernel 3: ex = exp(score - smax[dst]); score <- ex; denom[dst] += ex
// One thread per (edge, head).
// ---------------------------------------------------------------------------
__global__ void k_softmax_norm(const int* __restrict__ ei,
                               float* __restrict__ score,
                               const float* __restrict__ smax,
                               float* __restrict__ denom,
                               int E)
{
    const int tid = blockIdx.x * blockDim.x + threadIdx.x;
    if (tid >= E * NHEAD) return;
    const int e = tid >> 3;
    const int h = tid & (NHEAD - 1);
    const int dst = ei[E + e];
    const float ex = expf(score[tid] - smax[(size_t)dst * NHEAD + h]);
    score[tid] = ex;
    atomicAdd(&denom[(size_t)dst * NHEAD + h], ex);
}

// ---------------------------------------------------------------------------
// Kernel 4: agg[dst] += (ex/denom[dst]) * xl[src]. One wave per edge.
// ---------------------------------------------------------------------------
__global__ void k_aggregate(const int* __restrict__ ei,
                            const float* __restrict__ score,
                            const float* __restrict__ denom,
                            const float* __restrict__ xl,
                            float* __restrict__ agg,
                            int E)
{
    const int e = blockIdx.x * (blockDim.x >> 5) + (threadIdx.x >> 5);
    if (e >= E) return;
    const int lane = threadIdx.x & 31;
    const int src = ei[e];
    const int dst = ei[E + e];
    const int h = lane >> 2;                 // lane owns channels of head h
    const float alpha = score[(size_t)e * NHEAD + h] /
                        denom[(size_t)dst * NHEAD + h];
    const int c0 = lane * 4;
    #pragma unroll
    for (int j = 0; j < 4; j++) {
        const int c = c0 + j;
        atomicAdd(&agg[(size_t)dst * DMODEL + c],
                  alpha * xl[(size_t)src * DMODEL + c]);
    }
}

// ---------------------------------------------------------------------------
// Kernel 5: per-column sum / sumsq of (agg + bias) for BatchNorm stats.
// ---------------------------------------------------------------------------
__global__ void k_colstats(const float* __restrict__ agg,
                           const float* __restrict__ bias,
                           float* __restrict__ colsum,
                           float* __restrict__ colsumsq,
                           int n)
{
    const int c = threadIdx.x & (DMODEL - 1);
    const int rows_per_blk = blockDim.x / DMODEL;   // 2
    int r = blockIdx.x * rows_per_blk + (threadIdx.x / DMODEL);
    const int rstride = gridDim.x * rows_per_blk;
    const float bc = bias[c];
    float s = 0.0f, ss = 0.0f;
    for (; r < n; r += rstride) {
        const float v = agg[(size_t)r * DMODEL + c] + bc;
        s += v; ss += v * v;
    }
    atomicAdd(&colsum[c], s);
    atomicAdd(&colsumsq[c], ss);
}

// ---------------------------------------------------------------------------
// Kernel 6: BN (batch stats, biased var) + leaky_relu(0.01) -> d_out
// ---------------------------------------------------------------------------
__global__ void k_bn_out(const float* __restrict__ agg,
                         const float* __restrict__ bias,
                         const float* __restrict__ colsum,
                         const float* __restrict__ colsumsq,
                         const float* __restrict__ gamma,
                         const float* __restrict__ beta,
                         float* __restrict__ out,
                         int n)
{
    const size_t idx = (size_t)blockIdx.x * blockDim.x + threadIdx.x;
    if (idx >= (size_t)n * DMODEL) return;
    const int c = (int)(idx & (DMODEL - 1));
    const float invN = 1.0f / (float)n;
    const float mean = colsum[c] * invN;
    const float var  = colsumsq[c] * invN - mean * mean;
    const float v = agg[idx] + bias[c];
    float y = (v - mean) * rsqrtf(var + 1e-5f) * gamma[c] + beta[c];
    out[idx] = (y > 0.0f) ? y : 0.01f * y;
}

// ---------------------------------------------------------------------------
extern "C" void kernel_launch(void* const* d_in, const int* in_sizes, int n_in,
                              void* d_out, int out_size, void* d_ws, size_t ws_size,
                              hipStream_t stream) {
    const int*   x     = (const int*)  d_in[0];
    const int*   ei    = (const int*)  d_in[1];
    const float* ew    = (const float*)d_in[2];
    const float* emb   = (const float*)d_in[3];
    const float* Wl    = (const float*)d_in[4];
    const float* bl    = (const float*)d_in[5];
    const float* Wr    = (const float*)d_in[6];
    const float* br    = (const float*)d_in[7];
    const float* att   = (const float*)d_in[8];
    const float* We    = (const float*)d_in[9];
    const float* bias  = (const float*)d_in[10];
    const float* gamma = (const float*)d_in[11];
    const float* beta  = (const float*)d_in[12];
    float* out = (float*)d_out;

    const int N = in_sizes[0];
    const int E = in_sizes[1] / 2;

    // Workspace layout (floats):
    // [xl N*128][xr N*128][score E*8][smax N*8][denom N*8][agg N*128][colsum 128][colsumsq 128]
    float* ws       = (float*)d_ws;
    float* xl       = ws;
    float* xr       = xl    + (size_t)N * DMODEL;
    float* score    = xr    + (size_t)N * DMODEL;
    float* smax     = score + (size_t)E * NHEAD;
    float* denom    = smax  + (size_t)N * NHEAD;
    float* agg      = denom + (size_t)N * NHEAD;
    float* colsum   = agg   + (size_t)N * DMODEL;
    float* colsumsq = colsum + DMODEL;

    // Init (every call; ws is not re-poisoned but we fully re-init).
    const size_t zcount = (size_t)N * NHEAD + (size_t)N * DMODEL + 2 * DMODEL;
    k_fill<<<2048, 256, 0, stream>>>(smax, -__builtin_huge_valf(), (size_t)N * NHEAD);
    k_fill<<<2048, 256, 0, stream>>>(denom, 0.0f, zcount);   // denom+agg+colsum+colsumsq

    // 1) gather + dual GEMM (WMMA f32 16x16x4)
    k_embed_gemm<<<(N + 31) / 32, 256, 0, stream>>>(x, emb, Wl, bl, Wr, br, xl, xr, N);

    // 2) edge scores + segment max (one wave per edge)
    k_edge_score<<<(E + 7) / 8, 256, 0, stream>>>(ei, ew, We, att, xl, xr, score, smax, E);

    // 3) exp + segment sum
    k_softmax_norm<<<(E * NHEAD + 255) / 256, 256, 0, stream>>>(ei, score, smax, denom, E);

    // 4) weighted scatter-add aggregation
    k_aggregate<<<(E + 7) / 8, 256, 0, stream>>>(ei, score, denom, xl, agg, E);

    // 5) BN column stats
    k_colstats<<<512, 256, 0, stream>>>(agg, bias, colsum, colsumsq, N);

    // 6) BN normalize + leaky_relu -> out
    k_bn_out<<<(int)(((size_t)N * DMODEL + 255) / 256), 256, 0, stream>>>(
        agg, bias, colsum, colsumsq, gamma, beta, out, N);
}